// MVDRBeamformer_82085414961880
// MI455X (gfx1250) — compile-verified
//
#include <hip/hip_runtime.h>

typedef __attribute__((ext_vector_type(2))) float v2f;
typedef __attribute__((ext_vector_type(8))) float v8f;

#define EPS      1e-8f
#define PSD_EPS  1e-5f
#define B_   8
#define C_   8
#define F_   257
#define T_   800
#define BF_  (B_ * F_)          /* 2056 */
#define PSD_N (BF_ * 64)        /* floats per psd component plane */

// ---------------------------------------------------------------------------
// Kernel 1: masked complex Gram (PSD) via V_WMMA_F32_16X16X4_F32.
// ONE wave per (b,f) computes BOTH speech and noise PSDs so X is streamed
// once. Rows 0-7 of the 16xT A matrix carry Re(X*m), rows 8-15 carry
// Im(X*m); G = S*S^T gives all four real blocks of the complex Gram.
// Two independent accumulation chains (speech/noise) = matrix-pipe ILP.
// ---------------------------------------------------------------------------
__global__ __launch_bounds__(256) void psd_kernel(
    const float* __restrict__ smr, const float* __restrict__ smi,
    const float* __restrict__ nmr, const float* __restrict__ nmi,
    const float* __restrict__ xr,  const float* __restrict__ xi,
    float* __restrict__ ws)
{
    __shared__ float lds[8][2][16][16];       // two 16x16 tiles per wave (16 KB)

    const int lane  = threadIdx.x & 31;
    const int wslot = threadIdx.x >> 5;
    const int bf    = blockIdx.x * 8 + wslot; // 0..2055, exact (257 blocks)
    const int b = bf / F_;
    const int f = bf - b * F_;
    const int mbase = bf * T_;

    // den_s, den_n = sum_t |m|^2 for both masks (strided sum + butterfly)
    float den_s = 0.f, den_n = 0.f;
    #pragma unroll 5
    for (int t = lane; t < T_; t += 32) {
        float a = smr[mbase + t], bb = smi[mbase + t];
        float cN = nmr[mbase + t], dN = nmi[mbase + t];
        den_s += a * a + bb * bb;
        den_n += cN * cN + dN * dN;
    }
    for (int off = 16; off > 0; off >>= 1) {
        den_s += __shfl_xor(den_s, off, 32);
        den_n += __shfl_xor(den_n, off, 32);
    }

    // WMMA Gram accumulation. A-operand layout (32-bit, 16x4):
    //   lane m (0-15): row M=m at K = {k0, k0+1}; lanes 16-31 at K = {k0+2, k0+3}
    const int half    = lane >> 4;            // K sub-offset selector
    const int c       = lane & 7;             // channel of this row
    const int imagrow = (lane >> 3) & 1;      // rows 8-15 hold Im(sig)
    const int xbase   = ((b * C_ + c) * F_ + f) * T_;

    v8f acc_s = {0.f, 0.f, 0.f, 0.f, 0.f, 0.f, 0.f, 0.f};
    v8f acc_n = {0.f, 0.f, 0.f, 0.f, 0.f, 0.f, 0.f, 0.f};
    #pragma unroll 2
    for (int k0 = 0; k0 < T_; k0 += 4) {
        const int t0 = k0 + half * 2;
        float xr0 = xr[xbase + t0], xr1 = xr[xbase + t0 + 1];
        float xi0 = xi[xbase + t0], xi1 = xi[xbase + t0 + 1];

        float sr0 = smr[mbase + t0], sr1 = smr[mbase + t0 + 1];
        float si0 = smi[mbase + t0], si1 = smi[mbase + t0 + 1];
        float nr0 = nmr[mbase + t0], nr1 = nmr[mbase + t0 + 1];
        float ni0 = nmi[mbase + t0], ni1 = nmi[mbase + t0 + 1];

        v2f as, an;
        as.x = imagrow ? (xr0 * si0 + xi0 * sr0) : (xr0 * sr0 - xi0 * si0);
        as.y = imagrow ? (xr1 * si1 + xi1 * sr1) : (xr1 * sr1 - xi1 * si1);
        an.x = imagrow ? (xr0 * ni0 + xi0 * nr0) : (xr0 * nr0 - xi0 * ni0);
        an.y = imagrow ? (xr1 * ni1 + xi1 * nr1) : (xr1 * nr1 - xi1 * ni1);

        // D = A * A^T + C  (symmetric operand: same regs legal for A and B)
        acc_s = __builtin_amdgcn_wmma_f32_16x16x4_f32(
            false, as, false, as, (short)0, acc_s, false, false);
        acc_n = __builtin_amdgcn_wmma_f32_16x16x4_f32(
            false, an, false, an, (short)0, acc_n, false, false);
    }

    // Spill both 16x16 tiles to LDS: row = j + 8*half, col = lane&15
    #pragma unroll
    for (int j = 0; j < 8; ++j) {
        lds[wslot][0][j + 8 * half][lane & 15] = acc_s[j];
        lds[wslot][1][j + 8 * half][lane & 15] = acc_n[j];
    }
    __builtin_amdgcn_wave_barrier();   // same-wave LDS ordering fence

    // Extract complex PSDs: Re = G[c][d]+G[c+8][d+8], Im = G[c+8][d]-G[c][d+8]
    const float scale_s = 1.f / (den_s + EPS);
    const float scale_n = 1.f / (den_n + EPS);
    #pragma unroll
    for (int i = 0; i < 2; ++i) {
        int p  = lane + 32 * i;        // 64 entries: p = cc*8 + dd
        int cc = p >> 3, dd = p & 7;
        float re_s = lds[wslot][0][cc][dd]     + lds[wslot][0][cc + 8][dd + 8];
        float im_s = lds[wslot][0][cc + 8][dd] - lds[wslot][0][cc][dd + 8];
        float re_n = lds[wslot][1][cc][dd]     + lds[wslot][1][cc + 8][dd + 8];
        float im_n = lds[wslot][1][cc + 8][dd] - lds[wslot][1][cc][dd + 8];
        ws[0 * PSD_N + bf * 64 + p] = re_s * scale_s;
        ws[1 * PSD_N + bf * 64 + p] = im_s * scale_s + PSD_EPS;
        ws[2 * PSD_N + bf * 64 + p] = re_n * scale_n;
        ws[3 * PSD_N + bf * 64 + p] = im_n * scale_n + PSD_EPS;
    }
}

// ---------------------------------------------------------------------------
// Kernel 2: per-(b,f) 8x8 complex inverse (in-place Gauss-Jordan, Hermitian
// + eps*I so no pivoting), invE = inv + 1j*EPS, trace(invE@Sp), column 0,
// w = col0 / (tr + EPS). One problem per thread, matrix in a private LDS slice.
// ---------------------------------------------------------------------------
__global__ __launch_bounds__(64) void mvdr_weights_kernel(float* __restrict__ ws)
{
    __shared__ float2 Mats[64][64];    // 32 KB: 64 threads x (8x8 complex)
    const int tid = blockIdx.x * 64 + threadIdx.x;
    if (tid >= BF_) return;

    float2 (&A)[64] = Mats[threadIdx.x];
    const float* s_re = ws;
    const float* s_im = ws + PSD_N;
    const float* n_re = ws + 2 * PSD_N;
    const float* n_im = ws + 3 * PSD_N;
    const int base = tid * 64;

    // load noise PSD, add eye*EPS (real)
    for (int p = 0; p < 64; ++p) {
        float re = n_re[base + p], im = n_im[base + p];
        if ((p >> 3) == (p & 7)) re += EPS;
        A[p] = make_float2(re, im);
    }

    // in-place complex Gauss-Jordan inversion
    for (int k = 0; k < 8; ++k) {
        float2 pv = A[k * 8 + k];
        float d   = pv.x * pv.x + pv.y * pv.y;
        float ipr =  pv.x / d, ipi = -pv.y / d;
        for (int j = 0; j < 8; ++j) {
            if (j == k) continue;
            float2 v = A[k * 8 + j];
            A[k * 8 + j] = make_float2(v.x * ipr - v.y * ipi,
                                       v.x * ipi + v.y * ipr);
        }
        A[k * 8 + k] = make_float2(ipr, ipi);
        for (int i = 0; i < 8; ++i) {
            if (i == k) continue;
            float2 fv = A[i * 8 + k];
            for (int j = 0; j < 8; ++j) {
                if (j == k) continue;
                float2 kv  = A[k * 8 + j];
                float2 cur = A[i * 8 + j];
                cur.x -= fv.x * kv.x - fv.y * kv.y;
                cur.y -= fv.x * kv.y + fv.y * kv.x;
                A[i * 8 + j] = cur;
            }
            A[i * 8 + k] = make_float2(-(fv.x * ipr - fv.y * ipi),
                                       -(fv.x * ipi + fv.y * ipr));
        }
    }

    // invE = inv + 1j*EPS (elementwise, as in reference)
    for (int p = 0; p < 64; ++p) A[p].y += EPS;

    // tr = trace(invE @ Sp) + EPS
    float trr = 0.f, tri = 0.f;
    for (int i = 0; i < 8; ++i)
        for (int k = 0; k < 8; ++k) {
            float2 v = A[i * 8 + k];
            float spr = s_re[base + k * 8 + i], spi = s_im[base + k * 8 + i];
            trr += v.x * spr - v.y * spi;
            tri += v.x * spi + v.y * spr;
        }
    trr += EPS;

    float* w_re = ws + 4 * PSD_N;
    float* w_im = w_re + BF_ * 8;
    const float dtr = trr * trr + tri * tri;
    for (int i = 0; i < 8; ++i) {
        float cr = 0.f, ci = 0.f;
        for (int k = 0; k < 8; ++k) {
            float2 v = A[i * 8 + k];
            float spr = s_re[base + k * 8], spi = s_im[base + k * 8];
            cr += v.x * spr - v.y * spi;
            ci += v.x * spi + v.y * spr;
        }
        // complex divide (cr + i*ci) / (trr + i*tri)
        w_re[tid * 8 + i] = (cr * trr + ci * tri) / dtr;
        w_im[tid * 8 + i] = (ci * trr - cr * tri) / dtr;
    }
}

// ---------------------------------------------------------------------------
// Kernel 3: es[b,f,t] = sum_c conj(w[c]) * X[b,c,f,t]; out = [2,B,F,T].
// Two t-values per lane -> b64 loads/stores on the big streaming pass.
// ---------------------------------------------------------------------------
__global__ __launch_bounds__(256) void beamform_kernel(
    const float* __restrict__ xr, const float* __restrict__ xi,
    const float* __restrict__ ws, float* __restrict__ out)
{
    const int idx2 = blockIdx.x * 256 + threadIdx.x;   // pair index
    if (idx2 >= BF_ * (T_ / 2)) return;
    const int bf = idx2 / (T_ / 2);
    const int t  = (idx2 - bf * (T_ / 2)) * 2;
    const int b  = bf / F_;
    const int f  = bf - b * F_;

    const float* w_re = ws + 4 * PSD_N;
    const float* w_im = w_re + BF_ * 8;

    float er0 = 0.f, ei0 = 0.f, er1 = 0.f, ei1 = 0.f;
    #pragma unroll
    for (int c = 0; c < 8; ++c) {
        float wr = w_re[bf * 8 + c], wi = w_im[bf * 8 + c];
        int xo = ((b * C_ + c) * F_ + f) * T_ + t;
        float2 vr = *(const float2*)(xr + xo);
        float2 vi = *(const float2*)(xi + xo);
        er0 += wr * vr.x + wi * vi.x;    // Re(conj(w) * X)
        ei0 += wr * vi.x - wi * vr.x;    // Im(conj(w) * X)
        er1 += wr * vr.y + wi * vi.y;
        ei1 += wr * vi.y - wi * vr.y;
    }
    const int o = bf * T_ + t;
    *(float2*)(out + o)             = make_float2(er0, er1);
    *(float2*)(out + BF_ * T_ + o)  = make_float2(ei0, ei1);
}

// ---------------------------------------------------------------------------
extern "C" void kernel_launch(void* const* d_in, const int* in_sizes, int n_in,
                              void* d_out, int out_size, void* d_ws, size_t ws_size,
                              hipStream_t stream)
{
    const float* smr = (const float*)d_in[0];
    const float* smi = (const float*)d_in[1];
    const float* nmr = (const float*)d_in[2];
    const float* nmi = (const float*)d_in[3];
    const float* mxr = (const float*)d_in[4];
    const float* mxi = (const float*)d_in[5];
    float* ws  = (float*)d_ws;
    float* out = (float*)d_out;

    // 2056 waves (one per bf-pair, both masks), 8 waves/block -> 257 blocks exact
    psd_kernel<<<257, 256, 0, stream>>>(smr, smi, nmr, nmi, mxr, mxi, ws);

    mvdr_weights_kernel<<<(BF_ + 63) / 64, 64, 0, stream>>>(ws);

    // B*F*(T/2) = 822,400 pair-elements
    beamform_kernel<<<(BF_ * (T_ / 2) + 255) / 256, 256, 0, stream>>>(mxr, mxi, ws, out);
}